// ConvTreeBlockV2_14474039787886
// MI455X (gfx1250) — compile-verified
//
#include <hip/hip_runtime.h>
#include <hip/hip_bf16.h>

typedef __attribute__((ext_vector_type(2))) float v2f;
typedef __attribute__((ext_vector_type(8))) float v8f;

#define NPTS   1000000
#define KS2    9
#define CIN    4
#define COUT   16
#define BN_EPS 1e-5f
#define SLOPE  0.2f

#define TILES           (NPTS / 16)          // 62500 tiles of 16 points
#define WAVES_PER_BLOCK 4
#define NBLOCKS         (TILES / WAVES_PER_BLOCK)  // 15625, exact

__device__ __forceinline__ float leaky(float x) { return x >= 0.f ? x : SLOPE * x; }

// ---------------------------------------------------------------------------
// zero the BN statistics accumulators (64 floats) each call (graph-replayable)
// ---------------------------------------------------------------------------
__global__ void zero_stats_kernel(float* __restrict__ stats) {
    int i = threadIdx.x;
    if (i < 64) stats[i] = 0.f;
}

// ---------------------------------------------------------------------------
// conv1: out1[n,o] = sum_{j,c} data[ind[n,j],c] * w1[c,j,o] + b1[o]
// One wave = one 16-point tile; K=36 reduced as 9 WMMA f32 16x16x4 steps
// (one per neighbor j: the 4 input channels are one K-chunk).
// Also accumulates per-channel sum / sum-of-squares for BN1 (training stats).
// ---------------------------------------------------------------------------
__global__ __launch_bounds__(128) void conv1_kernel(
    const float* __restrict__ data, const int* __restrict__ ind,
    const float* __restrict__ w1, const float* __restrict__ b1,
    float* __restrict__ out1, float* __restrict__ stats)
{
    __shared__ float lsum[COUT];
    __shared__ float lsq[COUT];
    const int tid = threadIdx.x;
    if (tid < COUT) { lsum[tid] = 0.f; lsq[tid] = 0.f; }
    __syncthreads();

    const int lane = tid & 31;
    const int wave = tid >> 5;
    const int half = lane >> 4;      // 0: K even pair, 1: K odd pair (per ISA A/B layout)
    const int m    = lane & 15;      // matrix row (A) / col (B,C,D)
    const int tile = blockIdx.x * WAVES_PER_BLOCK + wave;
    const int row0 = tile * 16;
    const int point = row0 + m;

    // B fragments: chunk j covers K = j*4 + c ; B[k][o] = w1[c*KS2*COUT + j*COUT + o]
    v2f Bf[KS2];
#pragma unroll
    for (int j = 0; j < KS2; ++j) {
        Bf[j].x = w1[((2 * half + 0) * KS2 + j) * COUT + m];
        Bf[j].y = w1[((2 * half + 1) * KS2 + j) * COUT + m];
    }

    v8f acc = {};
#pragma unroll
    for (int j = 0; j < KS2; ++j) {
        const int idx = ind[point * KS2 + j];
        const float2 d = *reinterpret_cast<const float2*>(data + idx * CIN + 2 * half);
        v2f a; a.x = d.x; a.y = d.y;
        acc = __builtin_amdgcn_wmma_f32_16x16x4_f32(false, a, false, Bf[j],
                                                    (short)0, acc, false, false);
    }

    const float bias = b1[m];
    float psum = 0.f, psq = 0.f;
#pragma unroll
    for (int r = 0; r < 8; ++r) {
        float v = acc[r] + bias;
        acc[r] = v;
        psum += v; psq += v * v;
    }
#pragma unroll
    for (int r = 0; r < 8; ++r) {
        out1[(row0 + r + half * 8) * COUT + m] = acc[r];
    }

    atomicAdd(&lsum[m], psum);
    atomicAdd(&lsq[m],  psq);
    __syncthreads();
    if (tid < COUT) {
        atomicAdd(&stats[tid],        lsum[tid]);
        atomicAdd(&stats[COUT + tid], lsq[tid]);
    }
}

// ---------------------------------------------------------------------------
// finalize BN: scale = gamma * rsqrt(var+eps), shift = beta - mean*scale
// ---------------------------------------------------------------------------
__global__ void finalize_bn_kernel(const float* __restrict__ stats,
                                   const float* __restrict__ gamma,
                                   const float* __restrict__ beta,
                                   float* __restrict__ scale,
                                   float* __restrict__ shift)
{
    const int o = threadIdx.x;
    if (o < COUT) {
        const float inv_n = 1.0f / (float)NPTS;
        const float mean  = stats[o] * inv_n;
        const float var   = stats[COUT + o] * inv_n - mean * mean;
        const float s     = gamma[o] * rsqrtf(var + BN_EPS);
        scale[o] = s;
        shift[o] = beta[o] - mean * s;
    }
}

// ---------------------------------------------------------------------------
// conv2: gather out1[ind], apply BN1+leaky on the fly, GEMM vs w2 (K=144 as
// 36 WMMA chunks). Writes raw (pre-BN2) result to `raw2` (= d_out) and
// accumulates BN2 stats.
// ---------------------------------------------------------------------------
__global__ __launch_bounds__(128) void conv2_kernel(
    const float* __restrict__ out1, const int* __restrict__ ind,
    const float* __restrict__ w2, const float* __restrict__ scale1,
    const float* __restrict__ shift1, float* __restrict__ raw2,
    float* __restrict__ stats2)
{
    __shared__ float lsum[COUT];
    __shared__ float lsq[COUT];
    const int tid = threadIdx.x;
    if (tid < COUT) { lsum[tid] = 0.f; lsq[tid] = 0.f; }
    __syncthreads();

    const int lane = tid & 31;
    const int wave = tid >> 5;
    const int half = lane >> 4;
    const int m    = lane & 15;
    const int tile = blockIdx.x * WAVES_PER_BLOCK + wave;
    const int row0 = tile * 16;
    const int point = row0 + m;

    // BN1 coefficients for exactly the 8 channels this lane consumes
    float sc[8], sh[8];
#pragma unroll
    for (int t = 0; t < 4; ++t) {
#pragma unroll
        for (int p = 0; p < 2; ++p) {
            const int c = 4 * t + 2 * half + p;
            sc[2 * t + p] = scale1[c];
            sh[2 * t + p] = shift1[c];
        }
    }

    // B fragments for all 36 K-chunks: chunk(j,t) covers k = j*16 + 4t + {0..3}
    // B[k][o] = w2[c*KS2*COUT + j*COUT + o]
    v2f Bf[KS2 * 4];
#pragma unroll
    for (int j = 0; j < KS2; ++j) {
#pragma unroll
        for (int t = 0; t < 4; ++t) {
            const int c0 = 4 * t + 2 * half;
            Bf[j * 4 + t].x = w2[((c0 + 0) * KS2 + j) * COUT + m];
            Bf[j * 4 + t].y = w2[((c0 + 1) * KS2 + j) * COUT + m];
        }
    }

    v8f acc = {};
#pragma unroll
    for (int j = 0; j < KS2; ++j) {
        const int idx = ind[point * KS2 + j];
        const float* fbase = out1 + idx * COUT + 2 * half;
#pragma unroll
        for (int t = 0; t < 4; ++t) {
            const float2 q = *reinterpret_cast<const float2*>(fbase + 4 * t);
            v2f a;
            a.x = leaky(q.x * sc[2 * t + 0] + sh[2 * t + 0]);
            a.y = leaky(q.y * sc[2 * t + 1] + sh[2 * t + 1]);
            acc = __builtin_amdgcn_wmma_f32_16x16x4_f32(false, a, false, Bf[j * 4 + t],
                                                        (short)0, acc, false, false);
        }
    }

    float psum = 0.f, psq = 0.f;
#pragma unroll
    for (int r = 0; r < 8; ++r) {
        const float v = acc[r];
        raw2[(row0 + r + half * 8) * COUT + m] = v;
        psum += v; psq += v * v;
    }

    atomicAdd(&lsum[m], psum);
    atomicAdd(&lsq[m],  psq);
    __syncthreads();
    if (tid < COUT) {
        atomicAdd(&stats2[tid],        lsum[tid]);
        atomicAdd(&stats2[COUT + tid], lsq[tid]);
    }
}

// ---------------------------------------------------------------------------
// residual: out = leaky( BN2(raw2) + data @ wd.T ), in place on d_out.
// BN2 is folded into the WMMA C accumulator; data@wd.T is one 16x16x4 WMMA.
// ---------------------------------------------------------------------------
__global__ __launch_bounds__(128) void residual_kernel(
    const float* __restrict__ data, const float* __restrict__ wd,
    const float* __restrict__ scale2, const float* __restrict__ shift2,
    float* __restrict__ out)
{
    const int tid  = threadIdx.x;
    const int lane = tid & 31;
    const int wave = tid >> 5;
    const int half = lane >> 4;
    const int m    = lane & 15;
    const int tile = blockIdx.x * WAVES_PER_BLOCK + wave;
    const int row0 = tile * 16;

    // B[k=c][o=m] = wd[m*CIN + c]
    v2f b;
    b.x = wd[m * CIN + 2 * half + 0];
    b.y = wd[m * CIN + 2 * half + 1];

    // A[m][c] = data[(row0+m)*CIN + c]
    const float2 dv = *reinterpret_cast<const float2*>(data + (row0 + m) * CIN + 2 * half);
    v2f a; a.x = dv.x; a.y = dv.y;

    // C = BN2-normalized raw conv2 output
    const float s2 = scale2[m];
    const float t2 = shift2[m];
    v8f c;
#pragma unroll
    for (int r = 0; r < 8; ++r) {
        c[r] = out[(row0 + r + half * 8) * COUT + m] * s2 + t2;
    }

    v8f d = __builtin_amdgcn_wmma_f32_16x16x4_f32(false, a, false, b,
                                                  (short)0, c, false, false);
#pragma unroll
    for (int r = 0; r < 8; ++r) {
        out[(row0 + r + half * 8) * COUT + m] = leaky(d[r]);
    }
}

// ---------------------------------------------------------------------------
// launch
// ---------------------------------------------------------------------------
extern "C" void kernel_launch(void* const* d_in, const int* in_sizes, int n_in,
                              void* d_out, int out_size, void* d_ws, size_t ws_size,
                              hipStream_t stream)
{
    const float* data = (const float*)d_in[0];
    const int*   ind  = (const int*)  d_in[1];
    const float* w1   = (const float*)d_in[2];
    const float* b1   = (const float*)d_in[3];
    const float* g1   = (const float*)d_in[4];
    const float* be1  = (const float*)d_in[5];
    const float* w2   = (const float*)d_in[6];
    const float* g2   = (const float*)d_in[7];
    const float* be2  = (const float*)d_in[8];
    const float* wd   = (const float*)d_in[9];

    float* ws    = (float*)d_ws;
    float* stats = ws;        // 64 floats: s1sum[16], s1sq[16], s2sum[16], s2sq[16]
    float* coef  = ws + 64;   // 64 floats: scale1, shift1, scale2, shift2
    float* out1  = ws + 128;  // N*COUT floats (64 MB) intermediate
    float* out   = (float*)d_out;

    zero_stats_kernel<<<1, 64, 0, stream>>>(stats);
    conv1_kernel<<<NBLOCKS, 128, 0, stream>>>(data, ind, w1, b1, out1, stats);
    finalize_bn_kernel<<<1, 32, 0, stream>>>(stats, g1, be1, coef, coef + 16);
    conv2_kernel<<<NBLOCKS, 128, 0, stream>>>(out1, ind, w2, coef, coef + 16, out, stats + 32);
    finalize_bn_kernel<<<1, 32, 0, stream>>>(stats + 32, g2, be2, coef + 32, coef + 48);
    residual_kernel<<<NBLOCKS, 128, 0, stream>>>(data, wd, coef + 32, coef + 48, out);
}